// EmbeddingLayer_7808250544915
// MI455X (gfx1250) — compile-verified
//
#include <hip/hip_runtime.h>
#include <hip/hip_bf16.h>

// Y[j, b, i] = bit (2047-b) of (2*x[i,j]+1),  Y: [2048, 2048, 64] f32 (1 GiB out).
// v = 2x+1 < 4096  =>  columns b < 2036 are identically zero; only the last 12
// embedding columns carry data. Pure store-bandwidth kernel:
// floor = 1 GiB / 23.3 TB/s ~= 46 us on MI455X. No FLOPs -> WMMA inapplicable.
//
// Structure: total threads = 2^21 == grid stride, trip count = 2^26/2^21 = 32
// (compile-time). Adding the stride only flips bits >= 21 of the flat float4
// index, so q, b, shift are LOOP-INVARIANT per thread; only j (+64/iter)
// varies. The zero/live split is hoisted out of the loop: zero threads run a
// pure unrolled stream of NT 128-bit stores (512 B per wave32 instruction).

typedef __attribute__((ext_vector_type(4))) float v4f;

namespace {
constexpr int      kE       = 2048;                              // seq == emb
constexpr int      kB       = 64;                                // batch
constexpr unsigned kThreads = 256;                               // 8 waves/block
constexpr unsigned kBlocks  = 8192;
constexpr unsigned kStride  = kThreads * kBlocks;                // 2^21 float4 slots
constexpr unsigned kN4      = (unsigned)kE * kE * (kB / 4);      // 2^26 float4 stores
constexpr unsigned kIters   = kN4 / kStride;                     // 32, exact
}

__global__ __launch_bounds__(kThreads)
void bit_expand_kernel(const int* __restrict__ x, float* __restrict__ out) {
    const unsigned base = blockIdx.x * kThreads + threadIdx.x;

    // Flat float4 index t = ((j*2048 + b)*16) + q ; this thread's t sequence is
    // base + k*2^21, so bits [20:0] (q, b) never change across iterations.
    const unsigned q  = base & 15u;
    const unsigned b  = (base >> 4) & (unsigned)(kE - 1);
    const int   shift = (kE - 1) - (int)b;       // bit position, fixed per thread

    v4f* __restrict__ out4 = (v4f*)out;

    if (shift >= 12) {
        // ---- Zero column (2036 of 2048): pure NT streaming-store loop ----
        const v4f zero = {0.0f, 0.0f, 0.0f, 0.0f};
        unsigned t = base;
        #pragma unroll
        for (unsigned k = 0; k < kIters; ++k) {
            __builtin_nontemporal_store(zero, out4 + t);
            t += kStride;
        }
    } else {
        // ---- Live column (12 of 2048): fixed shift, j advances by 64/iter.
        // x is 512 KB (L2-resident); 4 scalar gathers per 16 B stored, and this
        // path is only 0.6% of total traffic.
        const unsigned i0 = q * 4u;
        unsigned j = base >> 15;                 // j0 in [0, 63]
        unsigned t = base;
        #pragma unroll
        for (unsigned k = 0; k < kIters; ++k) {
            const int v0 = 2 * x[(i0 + 0u) * (unsigned)kE + j] + 1;
            const int v1 = 2 * x[(i0 + 1u) * (unsigned)kE + j] + 1;
            const int v2 = 2 * x[(i0 + 2u) * (unsigned)kE + j] + 1;
            const int v3 = 2 * x[(i0 + 3u) * (unsigned)kE + j] + 1;
            v4f val;
            val.x = (float)((v0 >> shift) & 1);
            val.y = (float)((v1 >> shift) & 1);
            val.z = (float)((v2 >> shift) & 1);
            val.w = (float)((v3 >> shift) & 1);
            __builtin_nontemporal_store(val, out4 + t);
            t += kStride;
            j += 64u;                            // == kStride >> 15
        }
    }
}

extern "C" void kernel_launch(void* const* d_in, const int* in_sizes, int n_in,
                              void* d_out, int out_size, void* d_ws, size_t ws_size,
                              hipStream_t stream) {
    const int* x   = (const int*)d_in[0];   // [64, 2048] int32 tokens
    float*     out = (float*)d_out;         // [2048, 2048, 64] f32

    (void)in_sizes; (void)n_in; (void)out_size; (void)d_ws; (void)ws_size;

    hipLaunchKernelGGL(bit_expand_kernel, dim3(kBlocks), dim3(kThreads), 0, stream,
                       x, out);
}